// Policy_QNN_54657753808901
// MI455X (gfx1250) — compile-verified
//
#include <hip/hip_runtime.h>

// ---------------------------------------------------------------------------
// QNN policy head, closed form.
//
// Math (exact, derived from the reference circuit):
//   RZ layers are diagonal -> no effect on Z-basis probabilities.
//   CNOT chain on a product state is a basis permutation ->
//     <Z_0> = z0, <Z_1> = z0*z1, <Z_2> = z0*z1*z2, <Z_3> = z0*z1*z2*z3
//   with per-qubit  z_k = cos(trx)*cos(x_k)*cos(x_{k+4}) - sin(trx)*sin(x_k)
//   (state RX(trx)*RY(x_{k+4})*RX(x_k)|0>).  theta_rz is provably unused.
//   out = softmax([t0,t1,t2,t3]),  t in [-1,1] so exp() cannot overflow.
//
// Roofline: 48 MB HBM traffic, ~2.06 us @ 23.3 TB/s -> memory bound.
// WMMA f32 16x16x4, samples-on-N formulation: A = all-ones 16x4 (layout-proof),
// B = per-sample exp values (4x16). Every C row holds the 16 column sums with
// N = lane, so the denominator readback is a single cndmask (no shuffles).
// ---------------------------------------------------------------------------

typedef __attribute__((ext_vector_type(2))) float v2f;
typedef __attribute__((ext_vector_type(8))) float v8f;

__global__ __launch_bounds__(256) void qnn_policy_kernel(
    const float* __restrict__ x,         // [n, 8]
    const float* __restrict__ theta_rx,  // [1]
    float* __restrict__ out,             // [n, 4]
    int n)
{
    const int gid = blockIdx.x * blockDim.x + threadIdx.x;
    const int idx = (gid < n) ? gid : (n - 1);   // clamp: keep EXEC all-ones for WMMA

    const float4* __restrict__ xv = reinterpret_cast<const float4*>(x);
    float4 xa = xv[idx * 2 + 0];   // RX angles x[:,0..3]
    float4 xb = xv[idx * 2 + 1];   // RY angles x[:,4..7]

    const float trx = theta_rx[0];
    float st, ct;
    __sincosf(trx, &st, &ct);

    float sa0, ca0, sa1, ca1, sa2, ca2, sa3, ca3;
    __sincosf(xa.x, &sa0, &ca0);
    __sincosf(xa.y, &sa1, &ca1);
    __sincosf(xa.z, &sa2, &ca2);
    __sincosf(xa.w, &sa3, &ca3);

    const float z0 = ct * ca0 * __cosf(xb.x) - st * sa0;
    const float z1 = ct * ca1 * __cosf(xb.y) - st * sa1;
    const float z2 = ct * ca2 * __cosf(xb.z) - st * sa2;
    const float z3 = ct * ca3 * __cosf(xb.w) - st * sa3;

    // prefix products = <Z_k>; all in [-1, 1]
    const float t0 = z0;
    const float t1 = t0 * z1;
    const float t2 = t1 * z2;
    const float t3 = t2 * z3;

    // softmax numerators (no max-shift needed: exp range [0.37, 2.72])
    const float e0 = __expf(t0);
    const float e1 = __expf(t1);
    const float e2 = __expf(t2);
    const float e3 = __expf(t3);

    // ---- softmax denominator: column sums via V_WMMA_F32_16X16X4_F32 ------
    float denom;
#if __has_builtin(__builtin_amdgcn_wmma_f32_16x16x4_f32)
    {
        const int lane = (int)(threadIdx.x & 31u);     // wave32
        const int lo   = lane & 15;
        const bool hi16 = (lane >= 16);

        // B layout (4x16 f32, KxN): V0 -> lanes 0-15 K=0 / lanes 16-31 K=2,
        //                           V1 -> lanes 0-15 K=1 / lanes 16-31 K=3,
        //                           N = lane % 16 (column = sample).
        // wmma #1 columns = samples of lanes 0..15; #2 = samples of lanes 16..31.
        const float p2 = __shfl(e2, lo);        // partner (lane-16) values
        const float p3 = __shfl(e3, lo);
        const float q0 = __shfl(e0, lo | 16);   // partner (lane+16) values
        const float q1 = __shfl(e1, lo | 16);

        v2f A, B1, B2;
        A.x  = 1.0f;            A.y  = 1.0f;    // all-ones 16x4 A: layout-proof
        B1.x = hi16 ? p2 : e0;  B1.y = hi16 ? p3 : e1;
        B2.x = hi16 ? e2 : q0;  B2.y = hi16 ? e3 : q1;

        v8f Z = {0.f, 0.f, 0.f, 0.f, 0.f, 0.f, 0.f, 0.f};
        v8f d1 = __builtin_amdgcn_wmma_f32_16x16x4_f32(
            false, A, false, B1, (short)0, Z, false, false);
        v8f d2 = __builtin_amdgcn_wmma_f32_16x16x4_f32(
            false, A, false, B2, (short)0, Z, false, false);

        // C/D layout, VGPR 0: lanes 0-15 = C[0][lane]  -> own denom (wmma #1),
        //                     lanes 16-31 = C[8][lane-16] -> own denom (wmma #2).
        denom = hi16 ? d2[0] : d1[0];
    }
#else
    denom = (e0 + e1) + (e2 + e3);
#endif

#if __has_builtin(__builtin_amdgcn_rcpf)
    const float inv = __builtin_amdgcn_rcpf(denom);   // v_rcp_f32, no div expansion
#else
    const float inv = 1.0f / denom;
#endif

    if (gid < n) {
        float4 o;
        o.x = e0 * inv;
        o.y = e1 * inv;
        o.z = e2 * inv;
        o.w = e3 * inv;
        reinterpret_cast<float4*>(out)[gid] = o;
    }
}

extern "C" void kernel_launch(void* const* d_in, const int* in_sizes, int n_in,
                              void* d_out, int out_size, void* d_ws, size_t ws_size,
                              hipStream_t stream) {
    const float* x         = (const float*)d_in[0];
    const float* theta_rx  = (const float*)d_in[1];
    // d_in[2] = theta_rz: provably unused (diagonal gates drop out of Z stats)
    float* out = (float*)d_out;

    const int n = in_sizes[0] / 8;
    const int threads = 256;                   // 8 wave32s per block
    const int blocks  = (n + threads - 1) / threads;
    qnn_policy_kernel<<<blocks, threads, 0, stream>>>(x, theta_rx, out, n);
    (void)d_ws; (void)ws_size; (void)out_size; (void)n_in;
}